// LogEig_876173328409
// MI455X (gfx1250) — compile-verified
//
#include <hip/hip_runtime.h>
#include <hip/hip_bf16.h>
#include <math.h>

// ---------------------------------------------------------------------------
// Batched SPD matrix logarithm via inverse scaling-and-squaring, all-GEMM,
// using CDNA5 f32 WMMA (v_wmma_f32_16x16x4_f32) on wave32.
//   A = X / ||X||_F              (eigenvalues in (0,1])
//   repeat k=5: A <- sqrt(A)     (coupled Newton-Schulz, 3 GEMMs / iter)
//   E = A - I;  p(E) = Horner Mercator series, 8 terms (1 GEMM / term)
//   logm(X) = 2^5 * p(E) + ln(||X||_F) * I
// ---------------------------------------------------------------------------

typedef float v2f __attribute__((ext_vector_type(2)));
typedef float v8f __attribute__((ext_vector_type(8)));

#define N64      64
#define LDL      68            // LDS row stride (floats): conflict-free A frags
#define NTHREADS 256           // 8 wave32 per workgroup, 1 workgroup per matrix
#define SQRT_LEVELS 5
#define NS_ITERS    14         // even -> ping-pong parity preserved per level
#define SERIES_M    8

// 64x64 * 64x64 -> 64x64, all operands in LDS, f32 WMMA 16x16x4.
// 16 output tiles, 8 waves, 2 tiles per wave. EXEC uniform across block.
__device__ __forceinline__ void mm64(float* __restrict__ C,
                                     const float* __restrict__ A,
                                     const float* __restrict__ B,
                                     int tid)
{
    const int lane  = tid & 31;
    const int wave  = tid >> 5;
    const int half  = lane >> 4;   // 0: lanes 0-15, 1: lanes 16-31
    const int l15   = lane & 15;

#pragma unroll
    for (int t = 0; t < 2; ++t) {
        const int tile = wave * 2 + t;      // 0..15
        const int ti   = tile >> 2;         // row-tile
        const int tj   = tile & 3;          // col-tile

        v8f acc = {};
        // A fragment base: row = ti*16 + l15, k offset 2*half (ISA 16x4 layout)
        const float* arow = A + (ti * 16 + l15) * LDL + 2 * half;
        // B fragment base: k row = 2*half, col = tj*16 + l15
        const float* bcol = B + (2 * half) * LDL + tj * 16 + l15;

#pragma unroll
        for (int kb = 0; kb < N64; kb += 4) {
            v2f a = *(const v2f*)(arow + kb);        // A[m][kb+2h], A[m][kb+2h+1]
            v2f b;
            b.x = bcol[kb * LDL];                    // B[kb+2h  ][n]
            b.y = bcol[kb * LDL + LDL];              // B[kb+2h+1][n]
            acc = __builtin_amdgcn_wmma_f32_16x16x4_f32(
                      false, a, false, b, (short)0, acc, false, false);
        }

        // C/D layout: VGPR r -> row r (lanes 0-15) / row r+8 (lanes 16-31)
        float* crow = C + (ti * 16 + 8 * half) * LDL + tj * 16 + l15;
#pragma unroll
        for (int r = 0; r < 8; ++r)
            crow[r * LDL] = acc[r];
    }
}

__global__ __launch_bounds__(NTHREADS)
void logm_iss_kernel(const float* __restrict__ X, float* __restrict__ out)
{
    __shared__ float Y0[N64 * LDL];
    __shared__ float Y1[N64 * LDL];
    __shared__ float Z0[N64 * LDL];
    __shared__ float Z1[N64 * LDL];
    __shared__ float Wb[N64 * LDL];
    __shared__ float red[8];
    __shared__ float cpar[2];   // [0] = 1/||X||_F, [1] = ln(||X||_F)

    const int tid = threadIdx.x;
    const int b   = blockIdx.x;
    const float* Xg = X + (size_t)b * (N64 * N64);
    float*       Og = out + (size_t)b * (N64 * N64);

    // ---- load X into LDS (Y0) + Frobenius-norm partial sums -------------
    float ss = 0.f;
    for (int e = tid; e < N64 * N64; e += NTHREADS) {
        float v = Xg[e];
        Y0[(e >> 6) * LDL + (e & 63)] = v;
        ss += v * v;
    }
#pragma unroll
    for (int off = 16; off > 0; off >>= 1)
        ss += __shfl_down(ss, off, 32);
    if ((tid & 31) == 0) red[tid >> 5] = ss;
    __syncthreads();
    if (tid == 0) {
        float tot = 0.f;
#pragma unroll
        for (int w = 0; w < 8; ++w) tot += red[w];
        float c = sqrtf(tot);
        cpar[0] = 1.0f / c;
        cpar[1] = logf(c);
    }
    __syncthreads();

    // ---- scale: A = X / ||X||_F  (in Y0) --------------------------------
    {
        const float inv_c = cpar[0];
        for (int e = tid; e < N64 * N64; e += NTHREADS)
            Y0[(e >> 6) * LDL + (e & 63)] *= inv_c;
    }
    __syncthreads();

    // ---- k repeated square roots via coupled Newton-Schulz --------------
    // Y ping-pongs between Y0/Y1, Z between Z0/Z1. NS_ITERS even keeps the
    // current matrix in buffer `s` at the end of each level.
    int s = 0;
    for (int lvl = 0; lvl < SQRT_LEVELS; ++lvl) {
        float* Zs_init = s ? Z1 : Z0;
        for (int e = tid; e < N64 * N64; e += NTHREADS) {
            int r = e >> 6, c = e & 63;
            Zs_init[r * LDL + c] = (r == c) ? 1.0f : 0.0f;
        }
        __syncthreads();

        for (int it = 0; it < NS_ITERS; ++it) {
            float* Ys = s ? Y1 : Y0;
            float* Yd = s ? Y0 : Y1;
            float* Zs = s ? Z1 : Z0;
            float* Zd = s ? Z0 : Z1;

            mm64(Wb, Zs, Ys, tid);                    // W = Z*Y
            __syncthreads();
            for (int e = tid; e < N64 * N64; e += NTHREADS) {
                int r = e >> 6, c = e & 63;
                float w = Wb[r * LDL + c];
                Wb[r * LDL + c] = ((r == c) ? 1.5f : 0.0f) - 0.5f * w;  // T
            }
            __syncthreads();
            mm64(Yd, Ys, Wb, tid);                    // Y' = Y*T
            mm64(Zd, Wb, Zs, tid);                    // Z' = T*Z
            __syncthreads();
            s ^= 1;
        }
    }

    // ---- Mercator series: log(A) ~ p(E), E = A - I ----------------------
    float* Acur = s ? Y1 : Y0;
    for (int e = tid; e < N64 * N64; e += NTHREADS) {
        int r = e >> 6, c = e & 63;
        Z0[r * LDL + c] = Acur[r * LDL + c] - ((r == c) ? 1.0f : 0.0f); // E
        // Horner seed: H = c_m I, c_m = (-1)^(m+1)/m, m=8 -> -1/8
        Z1[r * LDL + c] = (r == c) ? (-1.0f / (float)SERIES_M) : 0.0f;
    }
    __syncthreads();

    for (int j = SERIES_M - 1; j >= 1; --j) {
        mm64(Wb, Z0, Z1, tid);                        // W = E*H
        __syncthreads();
        const float cj = ((j & 1) ? 1.0f : -1.0f) / (float)j;
        for (int e = tid; e < N64 * N64; e += NTHREADS) {
            int r = e >> 6, c = e & 63;
            Z1[r * LDL + c] = Wb[r * LDL + c] + ((r == c) ? cj : 0.0f);
        }
        __syncthreads();
    }
    mm64(Wb, Z0, Z1, tid);                            // W = E*H_1 = p(E)
    __syncthreads();

    // ---- logm(X) = 2^k * p(E) + ln(c) * I -------------------------------
    const float lc = cpar[1];
    const float scal = (float)(1 << SQRT_LEVELS);     // 32
    for (int e = tid; e < N64 * N64; e += NTHREADS) {
        int r = e >> 6, c = e & 63;
        Og[e] = scal * Wb[r * LDL + c] + ((r == c) ? lc : 0.0f);
    }
}

extern "C" void kernel_launch(void* const* d_in, const int* in_sizes, int n_in,
                              void* d_out, int out_size, void* d_ws, size_t ws_size,
                              hipStream_t stream)
{
    (void)n_in; (void)d_ws; (void)ws_size; (void)out_size;
    const float* X = (const float*)d_in[0];
    float* out = (float*)d_out;
    const int B = in_sizes[0] / (N64 * N64);          // 8192
    logm_iss_kernel<<<B, NTHREADS, 0, stream>>>(X, out);
}